// Model_14121852470215
// MI455X (gfx1250) — compile-verified
//
#include <hip/hip_runtime.h>
#include <hip/hip_bf16.h>
#include <math.h>

typedef __attribute__((ext_vector_type(2))) float v2f;
typedef __attribute__((ext_vector_type(8))) float v8f;

#define DT    0.01f
#define BETA  0.99f
#define THR   0.2f
#define QDEC  0.9f
#define TWOPI 6.283185307179586476925f

#define WMMA_F32(A_, B_, C_) \
  __builtin_amdgcn_wmma_f32_16x16x4_f32(false, (A_), false, (B_), (short)0, (C_), false, false)

// fp32 WMMA GEMM: Out[M,N] = A[M,K] @ B^T, B given as [N,K] row-major.
// Each wave computes a 32x64 macro-tile: 2 M-subtiles x 4 N-subtiles,
// 8 v8f accumulators. Double-buffered fragment prefetch; 32-bit offsets
// from uniform bases so loads use the saddr (GVS) form.
// Requires M % 32 == 0, N % 64 == 0, K % 4 == 0, buffers < 2^31 bytes.
__global__ __launch_bounds__(256) void gemm_wmma_f32(
    const float* __restrict__ A,     // [M, K] row-major
    const float* __restrict__ Brows, // [N, K] row-major (B transposed)
    float* __restrict__ Out,         // [M, N] row-major
    int M, int N, int K)
{
  const int wave  = threadIdx.x >> 5;
  const int lane  = threadIdx.x & 31;
  const int macro = blockIdx.x * 8 + wave;
  const int ntn   = N >> 6;                 // macro tiles along N (64 cols)
  const int ntot  = (M >> 5) * ntn;         // macro tiles total
  if (macro >= ntot) return;                // uniform per wave -> EXEC all-1s

  const int tm   = macro / ntn;
  const int tn   = macro % ntn;
  const int mn   = lane & 15;               // A row-in-subtile / B col-in-subtile
  const int half = lane >> 4;               // selects K pair {0,1} vs {2,3}

  // 32-bit element offsets from uniform bases (saddr-form loads)
  const int aoff0 = (tm * 32 + mn) * K + half * 2;
  const int aoff1 = aoff0 + 16 * K;
  const int boff  = (tn * 64 + mn) * K + half * 2;
  const int bstep = 16 * K;                 // stride between B N-subtiles

  v8f acc[2][4] = {};

  // Preload k = 0 fragments
  v2f a0 = *(const v2f*)(A + aoff0);
  v2f a1 = *(const v2f*)(A + aoff1);
  v2f b0 = *(const v2f*)(Brows + boff);
  v2f b1 = *(const v2f*)(Brows + boff + bstep);
  v2f b2 = *(const v2f*)(Brows + boff + 2 * bstep);
  v2f b3 = *(const v2f*)(Brows + boff + 3 * bstep);

  for (int k = 0; k < K; k += 4) {
    const int kn = (k + 4 < K) ? (k + 4) : k;   // clamped prefetch index (scalar)
    v2f na0 = *(const v2f*)(A + aoff0 + kn);
    v2f na1 = *(const v2f*)(A + aoff1 + kn);
    v2f nb0 = *(const v2f*)(Brows + boff + kn);
    v2f nb1 = *(const v2f*)(Brows + boff + bstep + kn);
    v2f nb2 = *(const v2f*)(Brows + boff + 2 * bstep + kn);
    v2f nb3 = *(const v2f*)(Brows + boff + 3 * bstep + kn);

    acc[0][0] = WMMA_F32(a0, b0, acc[0][0]);
    acc[0][1] = WMMA_F32(a0, b1, acc[0][1]);
    acc[0][2] = WMMA_F32(a0, b2, acc[0][2]);
    acc[0][3] = WMMA_F32(a0, b3, acc[0][3]);
    acc[1][0] = WMMA_F32(a1, b0, acc[1][0]);
    acc[1][1] = WMMA_F32(a1, b1, acc[1][1]);
    acc[1][2] = WMMA_F32(a1, b2, acc[1][2]);
    acc[1][3] = WMMA_F32(a1, b3, acc[1][3]);

    a0 = na0; a1 = na1; b0 = nb0; b1 = nb1; b2 = nb2; b3 = nb3;
  }

  // C/D layout: VGPR r holds row M = tileRow + 8*half + r, col N = tileCol + (lane&15)
#pragma unroll
  for (int i = 0; i < 2; ++i) {
#pragma unroll
    for (int j = 0; j < 4; ++j) {
      const int obase = (tm * 32 + i * 16 + half * 8) * N + tn * 64 + j * 16 + mn;
#pragma unroll
      for (int r = 0; r < 8; ++r)
        Out[obase + r * N] = acc[i][j][r];
    }
  }
}

// Per-(b,h) BRF resonator recurrence. u_io initially holds in_sum[b,t,h];
// overwritten in place with u. z/v/q written alongside. Stride-1 in h (coalesced).
__global__ void brf_recurrence(
    const float* __restrict__ omega,
    const float* __restrict__ boff,
    float* __restrict__ z_out,
    float* __restrict__ u_io,
    float* __restrict__ v_out,
    float* __restrict__ q_out,
    int bs, int seq, int dh)
{
  int idx = blockIdx.x * blockDim.x + threadIdx.x;
  if (idx >= bs * dh) return;
  const int b = idx / dh;
  const int h = idx - b * dh;

  const float om   = fabsf(omega[h] * TWOPI);
  const float dtom = DT * om;
  const float p_om = (-1.0f + sqrtf(1.0f - dtom * dtom)) / DT;
  const float b0   = p_om - fabsf(boff[h]);

  float u = 0.0f, v = 0.0f, q = 0.0f;
  size_t base = (size_t)b * seq * dh + h;
  for (int t = 0; t < seq; ++t) {
    const size_t o = base + (size_t)t * dh;
    const float insum = u_io[o];
    const float bc = b0 - q;
    const float u_ = fmaf(DT, fmaf(bc, u, fmaf(-om, v, insum)), u);
    const float v_ = fmaf(DT, fmaf(om, u, bc * v), v);
    const float z  = (u_ - 1.0f - q) > 0.0f ? 1.0f : 0.0f;
    const float q_ = fmaf(QDEC, q, z);
    z_out[o] = z;
    u_io[o]  = u_;
    v_out[o] = v_;
    q_out[o] = q_;
    u = u_; v = v_; q = q_;
  }
}

// Per-(b,o) leaky membrane recurrence. mem_io initially holds cur[b,t,o];
// overwritten in place with mem. spk written alongside.
__global__ void leaky_recurrence(
    float* __restrict__ spk_out,
    float* __restrict__ mem_io,
    int bs, int seq, int dout)
{
  int idx = blockIdx.x * blockDim.x + threadIdx.x;
  if (idx >= bs * dout) return;
  const int b = idx / dout;
  const int o = idx - b * dout;

  float mem = 0.0f;
  size_t base = (size_t)b * seq * dout + o;
  for (int t = 0; t < seq; ++t) {
    const size_t i = base + (size_t)t * dout;
    const float cur   = mem_io[i];
    const float reset = (mem > THR) ? 1.0f : 0.0f;
    const float mem_  = fmaf(BETA, mem, cur) - reset * THR;
    const float spk   = (mem_ > THR) ? 1.0f : 0.0f;
    spk_out[i] = spk;
    mem_io[i]  = mem_;
    mem = mem_;
  }
}

extern "C" void kernel_launch(void* const* d_in, const int* in_sizes, int n_in,
                              void* d_out, int out_size, void* d_ws, size_t ws_size,
                              hipStream_t stream) {
  const float* x     = (const float*)d_in[0]; // [32,512,512]
  const float* W_brf = (const float*)d_in[1]; // [1024,512]
  const float* omega = (const float*)d_in[2]; // [1024]
  const float* boff  = (const float*)d_in[3]; // [1024]
  const float* W_fc  = (const float*)d_in[4]; // [64,1024]
  float* out = (float*)d_out;

  const int bs = 32, seq = 512, din = 512, dh = 1024, dno = 64;
  const size_t n_small = (size_t)bs * seq * dno;  // 1,048,576
  const size_t n_big   = (size_t)bs * seq * dh;   // 16,777,216

  float* spk = out;
  float* mem = out + n_small;
  float* z   = out + 2 * n_small;
  float* u   = z + n_big;
  float* v   = u + n_big;
  float* q   = v + n_big;

  const int M = bs * seq; // 16384

  // Phase 1: in_sum = X @ W_brf^T -> staged in u region
  {
    int macros = (M / 32) * (dh / 64);  // 512 * 16 = 8192, divisible by 8
    gemm_wmma_f32<<<macros / 8, 256, 0, stream>>>(x, W_brf, u, M, dh, din);
  }
  // Phase 2: BRF resonator scan; writes z/v/q, rewrites u in place
  {
    int n = bs * dh;                    // 32768
    brf_recurrence<<<(n + 255) / 256, 256, 0, stream>>>(
        omega, boff, z, u, v, q, bs, seq, dh);
  }
  // Phase 3: cur = Z @ W_fc^T -> staged in mem region
  {
    int macros = (M / 32) * (dno / 64); // 512 * 1 = 512, divisible by 8
    gemm_wmma_f32<<<macros / 8, 256, 0, stream>>>(z, W_fc, mem, M, dno, dh);
  }
  // Phase 4: leaky membrane scan; writes spk, rewrites mem in place
  {
    int n = bs * dno;                   // 2048
    leaky_recurrence<<<(n + 127) / 128, 128, 0, stream>>>(
        spk, mem, bs, seq, dno);
  }
}